// KnowledgeLayer_6047313953252
// MI455X (gfx1250) — compile-verified
//
#include <hip/hip_runtime.h>
#include <math.h>

// Probabilistic-circuit forward pass (KnowledgeLayer):
//   encode -> product(gather4,+) -> sum(gather4, logsumexp) -> product -> sum
// Memory-bound: 128 MB of streamed int32 indices (HBM) + random gathers into
// 8 MB arrays (held in the 192 MB L2). Strategy:
//   * b128 non-temporal loads for the index streams (stream once, don't evict
//     the L2-resident gather arrays)           -> global_load_b128 ... th:NT
//   * global_prefetch_b8 one grid-stride ahead -> hides streamed-load latency
//   * wave32 blocks of 256 (8 waves/block)
// WMMA/SWMMAC intentionally unused: fan-in-4 sparse gather-reduce has no
// dense 16x16 tile structure; matrix ops would add work, not remove it.

typedef int v4i __attribute__((ext_vector_type(4)));

#define LOG2F_CONST 0.6931471805599453f
#define EPSF 1e-15f

__device__ __forceinline__ float log1mexp_dev(float x) {
    // log(1 - exp(x)) for x <= 0 (Maechler 2012), matches the reference branch
    if (x > -LOG2F_CONST) {
        return logf(-expm1f(x));
    } else {
        return log1pf(-expf(x));
    }
}

// xe[0] = -inf, xe[1] = 0, xe[2+2i] = pos[i], xe[3+2i] = log1mexp(pos[i])
__global__ void kl_encode_kernel(const float* __restrict__ pos,
                                 float* __restrict__ xe, int n) {
    int i = blockIdx.x * blockDim.x + threadIdx.x;
    if (i == 0) {
        xe[0] = -__builtin_inff();
        xe[1] = 0.0f;
    }
    if (i < n) {
        float p = pos[i];
        xe[2 + 2 * i] = p;
        xe[3 + 2 * i] = log1mexp_dev(p);
    }
}

// dst[i] = sum_{k<4} src[ptrs[4i+k]]   (product of probabilities in log space)
__global__ void kl_product_kernel(const float* __restrict__ src,
                                  const int* __restrict__ ptrs,
                                  float* __restrict__ dst, int m) {
    const v4i* __restrict__ pv = (const v4i*)ptrs;
    int stride = gridDim.x * blockDim.x;
    for (int i = blockIdx.x * blockDim.x + threadIdx.x; i < m; i += stride) {
        // Prefetch next iteration's index quad (streamed, no reuse).
        if (i + stride < m) __builtin_prefetch(pv + (i + stride), 0, 1);
        // Non-temporal b128: stream the index quad, keep L2 for gather arrays.
        v4i p = __builtin_nontemporal_load(pv + i);
        float a = src[p.x];
        float b = src[p.y];
        float c = src[p.z];
        float d = src[p.w];
        dst[i] = ((a + b) + c) + d;  // left-assoc like a sequential segment_sum
    }
}

// Segmented (fan-in 4) logsumexp with detached max, NaN->0, +EPSILON floor.
__global__ void kl_sum_kernel(const float* __restrict__ src,
                              const int* __restrict__ ptrs,
                              float* __restrict__ dst, int m) {
    const v4i* __restrict__ pv = (const v4i*)ptrs;
    int stride = gridDim.x * blockDim.x;
    for (int i = blockIdx.x * blockDim.x + threadIdx.x; i < m; i += stride) {
        if (i + stride < m) __builtin_prefetch(pv + (i + stride), 0, 1);
        v4i p = __builtin_nontemporal_load(pv + i);
        float g0 = src[p.x];
        float g1 = src[p.y];
        float g2 = src[p.z];
        float g3 = src[p.w];
        float mx = fmaxf(fmaxf(g0, g1), fmaxf(g2, g3));
        // exp(g - mx); if mx == -inf then g - mx = NaN -> nan_to_num -> 0
        float e0 = expf(g0 - mx);
        float e1 = expf(g1 - mx);
        float e2 = expf(g2 - mx);
        float e3 = expf(g3 - mx);
        e0 = isnan(e0) ? 0.0f : e0;
        e1 = isnan(e1) ? 0.0f : e1;
        e2 = isnan(e2) ? 0.0f : e2;
        e3 = isnan(e3) ? 0.0f : e3;
        float s = (((e0 + e1) + e2) + e3) + EPSF;
        dst[i] = logf(s) + mx;  // all -inf segment: log(EPS) + (-inf) = -inf
    }
}

extern "C" void kernel_launch(void* const* d_in, const int* in_sizes, int n_in,
                              void* d_out, int out_size, void* d_ws, size_t ws_size,
                              hipStream_t stream) {
    (void)n_in; (void)ws_size;
    const float* pos   = (const float*)d_in[0];
    const int*   ptrs0 = (const int*)d_in[1];
    const int*   ptrs1 = (const int*)d_in[2];
    const int*   ptrs2 = (const int*)d_in[3];
    const int*   ptrs3 = (const int*)d_in[4];
    // d_in[5] is csr = repeat(arange(M), 4): structure is implicit, unused.

    const int N = in_sizes[0];          // number of input variables
    const int E = in_sizes[1];          // edges per layer
    const int M = E / 4;                // nodes per layer (== out_size)
    (void)out_size;

    // Workspace layout: [xe: 2N+2 floats][bufA: M floats], 256B-aligned.
    size_t xe_bytes = (size_t)(2 * (size_t)N + 2) * sizeof(float);
    size_t xe_round = (xe_bytes + 255u) & ~(size_t)255u;
    float* xe   = (float*)d_ws;
    float* bufA = (float*)((char*)d_ws + xe_round);
    float* out  = (float*)d_out;

    const int TPB = 256;  // 8 wave32s per block
    // Encode: one thread per input variable.
    int enc_blocks = (N + TPB - 1) / TPB;
    hipLaunchKernelGGL(kl_encode_kernel, dim3(enc_blocks), dim3(TPB), 0, stream,
                       pos, xe, N);

    // Layers: grid-stride, ~2 nodes per thread so the prefetch has a target.
    int blocks = (M + 2 * TPB - 1) / (2 * TPB);
    if (blocks < 1) blocks = 1;

    // L1: product over encoded input
    hipLaunchKernelGGL(kl_product_kernel, dim3(blocks), dim3(TPB), 0, stream,
                       xe, ptrs0, bufA, M);
    // L2: sum layer -> d_out (used as ping buffer)
    hipLaunchKernelGGL(kl_sum_kernel, dim3(blocks), dim3(TPB), 0, stream,
                       bufA, ptrs1, out, M);
    // L3: product layer
    hipLaunchKernelGGL(kl_product_kernel, dim3(blocks), dim3(TPB), 0, stream,
                       out, ptrs2, bufA, M);
    // L4: sum layer -> final output
    hipLaunchKernelGGL(kl_sum_kernel, dim3(blocks), dim3(TPB), 0, stream,
                       bufA, ptrs3, out, M);
}